// Upsample_25056839205742
// MI455X (gfx1250) — compile-verified
//
#include <hip/hip_runtime.h>

typedef float v2f __attribute__((ext_vector_type(2)));
typedef float v8f __attribute__((ext_vector_type(8)));

#define CIN      384
#define COUT     192
#define NBATCH   4
#define NCOARSE  4096            // n per batch
#define NFINE    16384           // m per batch
#define ROWS_C   (NBATCH * NCOARSE)   // 16384
#define ROWS_F   (NBATCH * NFINE)     // 65536
#define PITCH_A  (CIN + 4)       // LDS row pitch (dwords): 388 % 64 == 4 -> conflict-free
#define PITCH_C  (COUT + 4)      // 196 % 64 == 4

// ---------------------------------------------------------------------------
// Kernel 1: h = LN(feats; ln2) @ w2 + b2      [16384 x 384] -> [16384 x 192]
// 128 threads = 4 waves; 16 rows per workgroup; each wave does 3 N-tiles of 16.
// ---------------------------------------------------------------------------
__global__ __launch_bounds__(128)
void k_ln_gemm2(const float* __restrict__ feats, const float* __restrict__ gamma,
                const float* __restrict__ beta, const float* __restrict__ w2,
                const float* __restrict__ b2, float* __restrict__ h) {
    __shared__ float xs[16 * PITCH_A];
    __shared__ float part[16][8];
    __shared__ float mu_s[16], rs_s[16];

    const int t = threadIdx.x;
    const int rowBase = blockIdx.x * 16;

    // cooperative load (float4), per-row into padded LDS
    for (int i = t; i < 16 * (CIN / 4); i += 128) {
        int r = i / (CIN / 4), c4 = i % (CIN / 4);
        ((float4*)(xs + r * PITCH_A))[c4] =
            ((const float4*)(feats + (size_t)(rowBase + r) * CIN))[c4];
    }
    __syncthreads();

    // LayerNorm: 8 threads per row
    {
        const int r = t >> 3, sub = t & 7;
        float s = 0.f, ss = 0.f;
        for (int c = sub; c < CIN; c += 8) {
            float v = xs[r * PITCH_A + c];
            s += v; ss += v * v;
        }
        part[r][sub] = s;
        __syncthreads();
        if (sub == 0) {
            float tot = 0.f;
            for (int k = 0; k < 8; ++k) tot += part[r][k];
            mu_s[r] = tot * (1.0f / CIN);
        }
        __syncthreads();
        part[r][sub] = ss;
        __syncthreads();
        if (sub == 0) {
            float tot = 0.f;
            for (int k = 0; k < 8; ++k) tot += part[r][k];
            float mu = mu_s[r];
            rs_s[r] = rsqrtf(tot * (1.0f / CIN) - mu * mu + 1e-5f);
        }
        __syncthreads();
        const float mu = mu_s[r], rs = rs_s[r];
        for (int c = sub; c < CIN; c += 8) {
            float v = xs[r * PITCH_A + c];
            xs[r * PITCH_A + c] = (v - mu) * rs * gamma[c] + beta[c];
        }
    }
    __syncthreads();

    // WMMA f32 16x16x4 over K=384
    const int wave = t >> 5, lane = t & 31;
    const int lo = lane & 15, hi = lane >> 4;
    for (int nt = wave; nt < COUT / 16; nt += 4) {
        const int nbase = nt * 16;
        v8f acc = {};
        for (int k = 0; k < CIN; k += 4) {
            const int ka = k + 2 * hi;
            v2f a, b;
            a.x = xs[lo * PITCH_A + ka];
            a.y = xs[lo * PITCH_A + ka + 1];
            b.x = w2[(size_t)ka * COUT + nbase + lo];
            b.y = w2[(size_t)(ka + 1) * COUT + nbase + lo];
            acc = __builtin_amdgcn_wmma_f32_16x16x4_f32(
                false, a, false, b, (short)0, acc, false, false);
        }
        const float bias = b2[nbase + lo];
        for (int r = 0; r < 8; ++r) {
            const int m = r + 8 * hi;
            h[(size_t)(rowBase + m) * COUT + nbase + lo] = acc[r] + bias;
        }
    }
}

// ---------------------------------------------------------------------------
// Kernel 2: per-batch 3-NN + inverse-distance weights.
// 256 threads, one fine point each; all 4096 coarse xyz staged in LDS (64KB).
// ---------------------------------------------------------------------------
__global__ __launch_bounds__(256)
void k_knn(const float* __restrict__ xyz, const float* __restrict__ sxyz,
           int* __restrict__ idx_out, float* __restrict__ w_out) {
    __shared__ float4 pts[NCOARSE];

    const int wgPerBatch = NFINE / 256;               // 64
    const int b = blockIdx.x / wgPerBatch;
    const int fineBase = b * NFINE + (blockIdx.x % wgPerBatch) * 256;
    const int coarseBase = b * NCOARSE;

    for (int i = threadIdx.x; i < NCOARSE; i += 256) {
        const float* p = xyz + (size_t)(coarseBase + i) * 3;
        pts[i] = make_float4(p[0], p[1], p[2], 0.f);
    }
    __syncthreads();

    const int f = fineBase + threadIdx.x;
    const float qx = sxyz[(size_t)f * 3 + 0];
    const float qy = sxyz[(size_t)f * 3 + 1];
    const float qz = sxyz[(size_t)f * 3 + 2];

    float b0 = 1e30f, b1 = 1e30f, b2 = 1e30f;
    int i0 = 0, i1 = 0, i2 = 0;
    for (int j = 0; j < NCOARSE; ++j) {
        const float4 p = pts[j];                       // LDS broadcast read
        const float dx = qx - p.x, dy = qy - p.y, dz = qz - p.z;
        const float d = dx * dx + dy * dy + dz * dz;
        if (d < b2) {                                  // strict '<' == top_k tie rule
            if (d < b1) {
                b2 = b1; i2 = i1;
                if (d < b0) { b1 = b0; i1 = i0; b0 = d; i0 = j; }
                else        { b1 = d;  i1 = j; }
            } else { b2 = d; i2 = j; }
        }
    }
    const float r0 = 1.f / (sqrtf(fmaxf(b0, 0.f)) + 1e-8f);
    const float r1 = 1.f / (sqrtf(fmaxf(b1, 0.f)) + 1e-8f);
    const float r2 = 1.f / (sqrtf(fmaxf(b2, 0.f)) + 1e-8f);
    const float inv = 1.f / (r0 + r1 + r2);
    idx_out[(size_t)f * 3 + 0] = coarseBase + i0;
    idx_out[(size_t)f * 3 + 1] = coarseBase + i1;
    idx_out[(size_t)f * 3 + 2] = coarseBase + i2;
    w_out[(size_t)f * 3 + 0] = r0 * inv;
    w_out[(size_t)f * 3 + 1] = r1 * inv;
    w_out[(size_t)f * 3 + 2] = r2 * inv;
}

// ---------------------------------------------------------------------------
// Kernel 3: out = LN(support_feats; ln1) @ w1 + b1 + interp(h)
// ---------------------------------------------------------------------------
__global__ __launch_bounds__(128)
void k_ln_gemm1_fused(const float* __restrict__ sfeats, const float* __restrict__ gamma,
                      const float* __restrict__ beta, const float* __restrict__ w1,
                      const float* __restrict__ b1, const float* __restrict__ h,
                      const int* __restrict__ knn_idx, const float* __restrict__ knn_w,
                      float* __restrict__ out) {
    __shared__ float xs[16 * PITCH_C];
    __shared__ float part[16][8];
    __shared__ float mu_s[16], rs_s[16];
    __shared__ int   idx_s[16][3];
    __shared__ float w_s[16][3];

    const int t = threadIdx.x;
    const int rowBase = blockIdx.x * 16;

    for (int i = t; i < 16 * (COUT / 4); i += 128) {
        int r = i / (COUT / 4), c4 = i % (COUT / 4);
        ((float4*)(xs + r * PITCH_C))[c4] =
            ((const float4*)(sfeats + (size_t)(rowBase + r) * COUT))[c4];
    }
    if (t < 48) {
        int r = t / 3, k = t % 3;
        idx_s[r][k] = knn_idx[(size_t)(rowBase + r) * 3 + k];
        w_s[r][k]   = knn_w[(size_t)(rowBase + r) * 3 + k];
    }
    __syncthreads();

    {
        const int r = t >> 3, sub = t & 7;
        float s = 0.f, ss = 0.f;
        for (int c = sub; c < COUT; c += 8) {
            float v = xs[r * PITCH_C + c];
            s += v; ss += v * v;
        }
        part[r][sub] = s;
        __syncthreads();
        if (sub == 0) {
            float tot = 0.f;
            for (int k = 0; k < 8; ++k) tot += part[r][k];
            mu_s[r] = tot * (1.0f / COUT);
        }
        __syncthreads();
        part[r][sub] = ss;
        __syncthreads();
        if (sub == 0) {
            float tot = 0.f;
            for (int k = 0; k < 8; ++k) tot += part[r][k];
            float mu = mu_s[r];
            rs_s[r] = rsqrtf(tot * (1.0f / COUT) - mu * mu + 1e-5f);
        }
        __syncthreads();
        const float mu = mu_s[r], rs = rs_s[r];
        for (int c = sub; c < COUT; c += 8) {
            float v = xs[r * PITCH_C + c];
            xs[r * PITCH_C + c] = (v - mu) * rs * gamma[c] + beta[c];
        }
    }
    __syncthreads();

    const int wave = t >> 5, lane = t & 31;
    const int lo = lane & 15, hi = lane >> 4;
    for (int nt = wave; nt < COUT / 16; nt += 4) {
        const int nbase = nt * 16;
        v8f acc = {};
        for (int k = 0; k < COUT; k += 4) {
            const int ka = k + 2 * hi;
            v2f a, b;
            a.x = xs[lo * PITCH_C + ka];
            a.y = xs[lo * PITCH_C + ka + 1];
            b.x = w1[(size_t)ka * COUT + nbase + lo];
            b.y = w1[(size_t)(ka + 1) * COUT + nbase + lo];
            acc = __builtin_amdgcn_wmma_f32_16x16x4_f32(
                false, a, false, b, (short)0, acc, false, false);
        }
        const float bias = b1[nbase + lo];
        const int col = nbase + lo;
        for (int r = 0; r < 8; ++r) {
            const int m = r + 8 * hi;
            float v = acc[r] + bias;
            v += w_s[m][0] * h[(size_t)idx_s[m][0] * COUT + col];
            v += w_s[m][1] * h[(size_t)idx_s[m][1] * COUT + col];
            v += w_s[m][2] * h[(size_t)idx_s[m][2] * COUT + col];
            out[(size_t)(rowBase + m) * COUT + col] = v;
        }
    }
}

// ---------------------------------------------------------------------------
// Kernel 4: pass-through tail outputs (support_xyz f32, support_offset i32 bits)
// ---------------------------------------------------------------------------
__global__ __launch_bounds__(256)
void k_tail(const float* __restrict__ sxyz, const int* __restrict__ soff,
            float* __restrict__ out_xyz, int nxyz) {
    const int i = blockIdx.x * 256 + threadIdx.x;
    if (i < nxyz) out_xyz[i] = sxyz[i];
    if (i < NBATCH) ((int*)(out_xyz + nxyz))[i] = soff[i];
}

extern "C" void kernel_launch(void* const* d_in, const int* in_sizes, int n_in,
                              void* d_out, int out_size, void* d_ws, size_t ws_size,
                              hipStream_t stream) {
    (void)in_sizes; (void)n_in; (void)out_size; (void)ws_size;
    const float* feats   = (const float*)d_in[0];   // [16384,384]
    const float* xyz     = (const float*)d_in[1];   // [16384,3]
    const float* sxyz    = (const float*)d_in[2];   // [65536,3]
    const float* sfeats  = (const float*)d_in[3];   // [65536,192]
    const int*   soff    = (const int*)d_in[5];     // [4]
    const float* ln1_g   = (const float*)d_in[6];
    const float* ln1_b   = (const float*)d_in[7];
    const float* w1      = (const float*)d_in[8];   // [192,192]
    const float* b1      = (const float*)d_in[9];
    const float* ln2_g   = (const float*)d_in[10];
    const float* ln2_b   = (const float*)d_in[11];
    const float* w2      = (const float*)d_in[12];  // [384,192]
    const float* b2      = (const float*)d_in[13];
    float* out = (float*)d_out;

    // workspace layout
    float* h     = (float*)d_ws;                               // 16384*192 f32
    int*   kidx  = (int*)((char*)d_ws + (size_t)ROWS_C * COUT * 4);
    float* kw    = (float*)(kidx + (size_t)ROWS_F * 3);

    // 1) h = LN2(feats) @ w2 + b2
    k_ln_gemm2<<<ROWS_C / 16, 128, 0, stream>>>(feats, ln2_g, ln2_b, w2, b2, h);
    // 2) 3-NN per batch
    k_knn<<<ROWS_F / 256, 256, 0, stream>>>(xyz, sxyz, kidx, kw);
    // 3) out = LN1(support_feats) @ w1 + b1 + interp
    k_ln_gemm1_fused<<<ROWS_F / 16, 128, 0, stream>>>(sfeats, ln1_g, ln1_b, w1, b1,
                                                      h, kidx, kw, out);
    // 4) tail: support_xyz + support_offset
    const int nxyz = ROWS_F * 3;
    k_tail<<<(nxyz + 255) / 256, 256, 0, stream>>>(sxyz, soff,
                                                   out + (size_t)ROWS_F * COUT, nxyz);
}